// GraphEncoder_90726889160783
// MI455X (gfx1250) — compile-verified
//
#include <hip/hip_runtime.h>
#include <math.h>

#define D_IN 256
#define HID  128
#define NEG_SLOPE 0.2f
#define EPS 1e-16f

typedef float v2f __attribute__((ext_vector_type(2)));
typedef float v8f __attribute__((ext_vector_type(8)));

// ---------- helpers ----------
__device__ __forceinline__ void atomicMaxF32(float* addr, float val) {
    // monotone float-max using integer atomics (works for mixed signs, init -inf)
    if (val >= 0.0f) atomicMax((int*)addr, __float_as_int(val));
    else             atomicMin((unsigned int*)addr, __float_as_uint(val));
}

__device__ __forceinline__ void atomAddF32(float* p, float v) {
    unsafeAtomicAdd(p, v);   // -> global_atomic_add_f32
}

__device__ __forceinline__ float leaky(float e) {
    return e > 0.0f ? e : NEG_SLOPE * e;
}

// ---------- 0) out = bias0 + bias1 ----------
__global__ void init_out_kernel(float* __restrict__ out, const float* __restrict__ bias, int total) {
    int i = blockIdx.x * blockDim.x + threadIdx.x;
    if (i < total) {
        int n = i & (HID - 1);
        out[i] = bias[n] + bias[HID + n];
    }
}

// ---------- 1) wt[k] = sum_n W[k][n] * att[n]  (block 0: src, block 1: dst) ----------
__global__ void wt_kernel(const float* __restrict__ Wsrc, const float* __restrict__ Wdst,
                          const float* __restrict__ att_src, const float* __restrict__ att_dst,
                          float* __restrict__ wt, int r) {
    const float* W = (blockIdx.x == 0) ? (Wsrc + (size_t)r * D_IN * HID)
                                       : (Wdst + (size_t)r * D_IN * HID);
    const float* a = (blockIdx.x == 0) ? (att_src + r * HID) : (att_dst + r * HID);
    float* o = wt + blockIdx.x * D_IN;
    int k = threadIdx.x;            // 256 threads
    float s = 0.0f;
    #pragma unroll 8
    for (int n = 0; n < HID; ++n) s += W[(size_t)k * HID + n] * a[n];
    o[k] = s;
}

// ---------- 2) h = x @ W  via V_WMMA_F32_16X16X4_F32 ----------
// Each wave: 64 rows x 16 cols (4 accumulator tiles). 8 waves/block -> 64 rows x 128 cols.
__global__ void __launch_bounds__(256) gemm_h_kernel(const float* __restrict__ x,
                                                     const float* __restrict__ W,
                                                     float* __restrict__ h, int nrows) {
    const int wave = threadIdx.x >> 5;
    const int lane = threadIdx.x & 31;
    const int l15  = lane & 15;
    const int half = lane >> 4;          // 0: K=0,1   1: K=2,3
    const int n0   = wave * 16;          // column tile
    const int m0   = blockIdx.x * 64;    // row base
    const int koff = half * 2;

    // clamp row reads for the final partial block (no EXEC divergence around WMMA)
    const int lim = nrows - 1;
    const size_t c0 = (size_t)min(m0 +  0 + l15, lim) * D_IN;
    const size_t c1 = (size_t)min(m0 + 16 + l15, lim) * D_IN;
    const size_t c2 = (size_t)min(m0 + 32 + l15, lim) * D_IN;
    const size_t c3 = (size_t)min(m0 + 48 + l15, lim) * D_IN;

    v8f A0 = {0,0,0,0,0,0,0,0};
    v8f A1 = A0, A2 = A0, A3 = A0;

    for (int k0 = 0; k0 < D_IN; k0 += 4) {
        const int kk = k0 + koff;
        v2f b;
        b.x = W[(size_t)kk * HID + n0 + l15];
        b.y = W[(size_t)(kk + 1) * HID + n0 + l15];
        v2f a0 = *(const v2f*)(x + c0 + kk);
        v2f a1 = *(const v2f*)(x + c1 + kk);
        v2f a2 = *(const v2f*)(x + c2 + kk);
        v2f a3 = *(const v2f*)(x + c3 + kk);
        A0 = __builtin_amdgcn_wmma_f32_16x16x4_f32(false, a0, false, b, (short)0, A0, false, false);
        A1 = __builtin_amdgcn_wmma_f32_16x16x4_f32(false, a1, false, b, (short)0, A1, false, false);
        A2 = __builtin_amdgcn_wmma_f32_16x16x4_f32(false, a2, false, b, (short)0, A2, false, false);
        A3 = __builtin_amdgcn_wmma_f32_16x16x4_f32(false, a3, false, b, (short)0, A3, false, false);
    }

    // C layout: VGPR j -> M = j (lanes 0-15) / j+8 (lanes 16-31), N = lane&15
    const int col = n0 + l15;
    #pragma unroll
    for (int j = 0; j < 8; ++j) {
        int row = m0 + j + half * 8;
        if (row      < nrows) h[(size_t)(row     ) * HID + col] = A0[j];
        if (row + 16 < nrows) h[(size_t)(row + 16) * HID + col] = A1[j];
        if (row + 32 < nrows) h[(size_t)(row + 32) * HID + col] = A2[j];
        if (row + 48 < nrows) h[(size_t)(row + 48) * HID + col] = A3[j];
    }
}

// ---------- 3) per-node: a_src, a_dst = x[i]·wt_src, x[i]·wt_dst; init m=-inf, den=0 ----------
__global__ void __launch_bounds__(256) anode_kernel(const float* __restrict__ x,
                                                    const float* __restrict__ wt,
                                                    float* __restrict__ a_s, float* __restrict__ a_d,
                                                    float* __restrict__ m,  float* __restrict__ den,
                                                    int N) {
    int node = (blockIdx.x * blockDim.x + threadIdx.x) >> 5;
    int lane = threadIdx.x & 31;
    if (node >= N) return;
    const float4* xr  = (const float4*)(x + (size_t)node * D_IN);
    const float4* ws4 = (const float4*)(wt);
    const float4* wd4 = (const float4*)(wt + D_IN);
    float4 p = xr[lane], q = xr[lane + 32];
    float4 s1 = ws4[lane], s2 = ws4[lane + 32];
    float4 d1 = wd4[lane], d2 = wd4[lane + 32];
    float ss = p.x*s1.x + p.y*s1.y + p.z*s1.z + p.w*s1.w
             + q.x*s2.x + q.y*s2.y + q.z*s2.z + q.w*s2.w;
    float dd = p.x*d1.x + p.y*d1.y + p.z*d1.z + p.w*d1.w
             + q.x*d2.x + q.y*d2.y + q.z*d2.z + q.w*d2.w;
    #pragma unroll
    for (int off = 16; off > 0; off >>= 1) {
        ss += __shfl_xor(ss, off, 32);
        dd += __shfl_xor(dd, off, 32);
    }
    if (lane == 0) {
        a_s[node] = ss;
        a_d[node] = dd;
        m[node]   = -INFINITY;
        den[node] = 0.0f;
    }
}

// ---------- 4) edge pass 1: segment max ----------
__global__ void edge_max_kernel(const int* __restrict__ ei,
                                const float* __restrict__ a_s, const float* __restrict__ a_d,
                                float* __restrict__ m, int E, int N) {
    int idx = blockIdx.x * blockDim.x + threadIdx.x;
    int total = E + N;
    if (idx >= total) return;
    int s, d;
    if (idx < E) { s = ei[idx]; d = ei[E + idx]; } else { s = idx - E; d = s; }
    float e = leaky(a_s[s] + a_d[d]);
    atomicMaxF32(&m[d], e);
}

// ---------- 5) edge pass 2: segment sum of exp ----------
__global__ void edge_sum_kernel(const int* __restrict__ ei,
                                const float* __restrict__ a_s, const float* __restrict__ a_d,
                                const float* __restrict__ m, float* __restrict__ den,
                                int E, int N) {
    int idx = blockIdx.x * blockDim.x + threadIdx.x;
    int total = E + N;
    if (idx >= total) return;
    int s, d;
    if (idx < E) { s = ei[idx]; d = ei[E + idx]; } else { s = idx - E; d = s; }
    float e = leaky(a_s[s] + a_d[d]);
    atomAddF32(&den[d], __expf(e - m[d]));
}

// ---------- 6) edge pass 3: out[dst] += alpha * h[src]  (one wave per edge) ----------
__global__ void __launch_bounds__(256) edge_agg_kernel(const int* __restrict__ ei,
                                                       const float* __restrict__ a_s,
                                                       const float* __restrict__ a_d,
                                                       const float* __restrict__ m,
                                                       const float* __restrict__ den,
                                                       const float* __restrict__ h,
                                                       float* __restrict__ out, int E, int N) {
    int gw   = (blockIdx.x * blockDim.x + threadIdx.x) >> 5;
    int lane = threadIdx.x & 31;
    int total = E + N;
    if (gw >= total) return;
    int s, d;
    if (gw < E) { s = ei[gw]; d = ei[E + gw]; } else { s = gw - E; d = s; }
    float e = leaky(a_s[s] + a_d[d]);
    float alpha = __expf(e - m[d]) / (den[d] + EPS);
    const float4* hv = (const float4*)(h + (size_t)s * HID);
    float4 v = hv[lane];                       // 32 lanes x 16B = full 512B row
    float* op = out + (size_t)d * HID + lane * 4;
    atomAddF32(op + 0, alpha * v.x);
    atomAddF32(op + 1, alpha * v.y);
    atomAddF32(op + 2, alpha * v.z);
    atomAddF32(op + 3, alpha * v.w);
}

// ---------- 7) ELU in place ----------
__global__ void elu_kernel(float* __restrict__ out, int total) {
    int i = blockIdx.x * blockDim.x + threadIdx.x;
    if (i < total) {
        float v = out[i];
        out[i] = v > 0.0f ? v : expm1f(v);
    }
}

extern "C" void kernel_launch(void* const* d_in, const int* in_sizes, int n_in,
                              void* d_out, int out_size, void* d_ws, size_t ws_size,
                              hipStream_t stream) {
    const float* x       = (const float*)d_in[0];
    const int*   ei0     = (const int*)  d_in[1];
    const int*   ei1     = (const int*)  d_in[2];
    const float* Wsrc    = (const float*)d_in[3];   // [2,256,128]
    const float* Wdst    = (const float*)d_in[4];   // [2,256,128]
    const float* att_src = (const float*)d_in[5];   // [2,128]
    const float* att_dst = (const float*)d_in[6];   // [2,128]
    const float* bias    = (const float*)d_in[7];   // [2,128]
    float* out = (float*)d_out;

    const int N = in_sizes[0] / D_IN;
    const int E = in_sizes[1] / 2;

    // workspace layout (floats): h[N*128] | a_s[N] | a_d[N] | m[N] | den[N] | wt[512]
    float* h   = (float*)d_ws;
    float* a_s = h   + (size_t)N * HID;
    float* a_d = a_s + N;
    float* m   = a_d + N;
    float* den = m   + N;
    float* wt  = den + N;

    const int total_out = N * HID;
    const int tb = 256;

    init_out_kernel<<<(total_out + tb - 1) / tb, tb, 0, stream>>>(out, bias, total_out);

    for (int r = 0; r < 2; ++r) {
        const int* ei = (r == 0) ? ei0 : ei1;
        int totalE = E + N;

        wt_kernel<<<2, 256, 0, stream>>>(Wsrc, Wdst, att_src, att_dst, wt, r);
        gemm_h_kernel<<<(N + 63) / 64, 256, 0, stream>>>(x, Wsrc + (size_t)r * D_IN * HID, h, N);
        anode_kernel<<<(N + 7) / 8, 256, 0, stream>>>(x, wt, a_s, a_d, m, den, N);
        edge_max_kernel<<<(totalE + tb - 1) / tb, tb, 0, stream>>>(ei, a_s, a_d, m, E, N);
        edge_sum_kernel<<<(totalE + tb - 1) / tb, tb, 0, stream>>>(ei, a_s, a_d, m, den, E, N);
        edge_agg_kernel<<<((size_t)totalE * 32 + tb - 1) / tb, tb, 0, stream>>>(
            ei, a_s, a_d, m, den, h, out, E, N);
    }

    elu_kernel<<<(total_out + tb - 1) / tb, tb, 0, stream>>>(out, total_out);
}